// Model_62440234549253
// MI455X (gfx1250) — compile-verified
//
#include <hip/hip_runtime.h>

#define NUM_USERS 100000
#define NUM_ITEMS 50000
#define NUM_NODES (NUM_USERS + NUM_ITEMS)
#define EMBED_DIM 64
#define N_LAYERS 3
#define N_EDGES 2000000
#define BATCH 16384

typedef __attribute__((ext_vector_type(2))) float v2f;
typedef __attribute__((ext_vector_type(8))) float v8f;

// ---------------- elementwise helpers ----------------

__global__ void zero_f4(float4* __restrict__ p, int n4) {
  int i = blockIdx.x * blockDim.x + threadIdx.x;
  if (i < n4) p[i] = make_float4(0.f, 0.f, 0.f, 0.f);
}

__global__ void init_x_acc(const float4* __restrict__ ue, const float4* __restrict__ ie,
                           float4* __restrict__ x, float4* __restrict__ acc,
                           int nU4, int n4) {
  int i = blockIdx.x * blockDim.x + threadIdx.x;
  if (i >= n4) return;
  float4 v = (i < nU4) ? ue[i] : ie[i - nU4];
  x[i] = v;
  acc[i] = v;
}

__global__ void acc_add_f4(float4* __restrict__ acc, const float4* __restrict__ x, int n4) {
  int i = blockIdx.x * blockDim.x + threadIdx.x;
  if (i >= n4) return;
  float4 a = acc[i];
  float4 b = x[i];
  a.x += b.x; a.y += b.y; a.z += b.z; a.w += b.w;
  acc[i] = a;
}

// ---------------- normalization ----------------

__global__ void degree_kernel(const int* __restrict__ col, float* __restrict__ deg, int E) {
  int e = blockIdx.x * blockDim.x + threadIdx.x;
  if (e < E) atomicAdd(&deg[col[e]], 1.0f);
}

__global__ void rsqrt_kernel(float* __restrict__ deg, int n) {
  int i = blockIdx.x * blockDim.x + threadIdx.x;
  if (i < n) {
    float d = deg[i];
    deg[i] = (d > 0.f) ? rsqrtf(fmaxf(d, 1.f)) : 0.f;
  }
}

__global__ void norm_kernel(const int* __restrict__ row, const int* __restrict__ col,
                            const float* __restrict__ dis, float* __restrict__ nrm, int E) {
  int e = blockIdx.x * blockDim.x + threadIdx.x;
  if (e < E) nrm[e] = dis[row[e]] * dis[col[e]];
}

// ---------------- gather-scale-scatter conv (wave per edge) ----------------
// One wave32 handles one edge: 32 lanes x float2 = 64 floats (full row).
// Edge id is wave-uniform; readfirstlane pushes row/col/norm loads onto the
// scalar path (s_load). Scatter uses non-returning global_atomic_add_f32,
// which resolves in L2 (entire feature matrix fits in the 192MB L2).
__global__ void conv_kernel(const float* __restrict__ x, float* __restrict__ xn,
                            const int* __restrict__ row, const int* __restrict__ col,
                            const float* __restrict__ nrm, int E) {
  int wv = blockIdx.x * (blockDim.x >> 5) + (threadIdx.x >> 5);
  if (wv >= E) return;
  int lane = threadIdx.x & 31;
  int e = __builtin_amdgcn_readfirstlane(wv);
  int r = row[e];
  int c = col[e];
  float w = nrm[e];
  const v2f* src = (const v2f*)(x + (size_t)r * EMBED_DIM);
  v2f v = src[lane];
  float* dst = xn + (size_t)c * EMBED_DIM + lane * 2;
  atomicAdd(dst + 0, w * v.x);
  atomicAdd(dst + 1, w * v.y);
}

// ---------------- batched dot products via WMMA ----------------
// One wave per 16 (user,item) pairs. S = U * I^T accumulated over K=64 with
// 16 chained V_WMMA_F32_16X16X4_F32; diagonal of S gives the 16 dots.
// A (16x4 f32): lane L holds A[L%16][k0 + 2*(L/16) + {0,1}] in 2 VGPRs.
// B (4x16 f32): lane L holds B[k0' + {0,1}][L%16] with k0' = k0 + 2*(L/16),
// and B[k][n] = I[n][k] -> identical float2 load pattern as A.
// C/D 16x16 f32: element (M = j + 8*(L/16), N = L%16) in VGPR j.
__global__ void predict_kernel(const float* __restrict__ acc,
                               const int* __restrict__ user_ids,
                               const int* __restrict__ item_ids,
                               float* __restrict__ out, int nTiles) {
  int tile = blockIdx.x * (blockDim.x >> 5) + (threadIdx.x >> 5);
  if (tile >= nTiles) return;
  int lane = threadIdx.x & 31;
  int m  = lane & 15;
  int hi = lane >> 4;           // 0: K pair {0,1}, 1: K pair {2,3}
  int base = tile * 16;

  int uid = user_ids[base + m];
  int iid = item_ids[base + m];
  const float* urow = acc + (size_t)uid * EMBED_DIM;
  const float* irow = acc + ((size_t)NUM_USERS + (size_t)iid) * EMBED_DIM;

  v8f d = {};
#pragma unroll
  for (int k0 = 0; k0 < EMBED_DIM; k0 += 4) {
    int kk = k0 + 2 * hi;
    v2f a = *(const v2f*)(urow + kk);
    v2f b = *(const v2f*)(irow + kk);
    d = __builtin_amdgcn_wmma_f32_16x16x4_f32(
        /*neg_a=*/false, a, /*neg_b=*/false, b,
        /*c_mod=*/(short)0, d, /*reuse_a=*/false, /*reuse_b=*/false);
  }

  // Extract diagonal: pair p<8 lives in lane p, VGPR p; p>=8 in lane 16+p, VGPR p-8.
  float dv = 0.f;
#pragma unroll
  for (int j = 0; j < 8; ++j) {
    if (lane == j)       dv = d[j];
    if (lane == 24 + j)  dv = d[j];
  }
  bool active = (lane < 8) || (lane >= 24);
  int p = (lane < 8) ? lane : (lane - 16);
  if (active) {
    // final = acc / (N_LAYERS+1) for both operands -> scale dot by 1/16
    float dot = dv * (1.0f / 16.0f);
    out[base + p] = 1.0f / (1.0f + __expf(-dot));
  }
}

// ---------------- host launcher ----------------

extern "C" void kernel_launch(void* const* d_in, const int* in_sizes, int n_in,
                              void* d_out, int out_size, void* d_ws, size_t ws_size,
                              hipStream_t stream) {
  (void)in_sizes; (void)n_in; (void)out_size; (void)ws_size;

  const float* user_emb = (const float*)d_in[0];
  const float* item_emb = (const float*)d_in[1];
  const int*   user_ids = (const int*)d_in[2];
  const int*   item_ids = (const int*)d_in[3];
  const int*   edge     = (const int*)d_in[4];
  const int*   row = edge;              // edge_index[0]
  const int*   col = edge + N_EDGES;    // edge_index[1]
  float* out = (float*)d_out;

  // Workspace layout (floats): deg/dis | norm | x_a | x_b | acc
  const size_t ND   = (size_t)NUM_NODES * EMBED_DIM;   // 9.6M floats
  const size_t nAl  = 150016;                          // NUM_NODES aligned to 64
  float* ws  = (float*)d_ws;
  float* deg = ws;                      // N floats (becomes deg_inv_sqrt)
  float* nrm = ws + nAl;                // E floats
  float* xa  = nrm + N_EDGES;           // ND floats
  float* xb  = xa + ND;                 // ND floats
  float* acc = xb + ND;                 // ND floats

  const int T = 256;
  const int n4   = (int)(ND / 4);                   // 2,400,000
  const int nU4  = (NUM_USERS * EMBED_DIM) / 4;     // 1,600,000
  const int deg4 = NUM_NODES / 4;                   // 37,500

  // 0) zero degree, build x0 = concat(user_emb, item_emb), acc = x0
  zero_f4<<<(deg4 + T - 1) / T, T, 0, stream>>>((float4*)deg, deg4);
  init_x_acc<<<(n4 + T - 1) / T, T, 0, stream>>>(
      (const float4*)user_emb, (const float4*)item_emb,
      (float4*)xa, (float4*)acc, nU4, n4);

  // 1) degrees over destination nodes, then rsqrt, then per-edge norm
  degree_kernel<<<(N_EDGES + T - 1) / T, T, 0, stream>>>(col, deg, N_EDGES);
  rsqrt_kernel<<<(NUM_NODES + T - 1) / T, T, 0, stream>>>(deg, NUM_NODES);
  norm_kernel<<<(N_EDGES + T - 1) / T, T, 0, stream>>>(row, col, deg, nrm, N_EDGES);

  // 2) three LightGCN propagation layers, acc += x each layer
  const int wavesPerBlock = T / 32;                               // 8
  const int convBlocks = (N_EDGES + wavesPerBlock - 1) / wavesPerBlock;  // 250,000
  float* cur = xa;
  float* nxt = xb;
  for (int l = 0; l < N_LAYERS; ++l) {
    zero_f4<<<(n4 + T - 1) / T, T, 0, stream>>>((float4*)nxt, n4);
    conv_kernel<<<convBlocks, T, 0, stream>>>(cur, nxt, row, col, nrm, N_EDGES);
    acc_add_f4<<<(n4 + T - 1) / T, T, 0, stream>>>((float4*)acc, (const float4*)nxt, n4);
    float* t = cur; cur = nxt; nxt = t;
  }

  // 3) batched scoring: 1024 tiles of 16 pairs, WMMA f32 16x16x4
  const int nTiles = BATCH / 16;                                  // 1024
  predict_kernel<<<(nTiles + wavesPerBlock - 1) / wavesPerBlock, T, 0, stream>>>(
      acc, user_ids, item_ids, out, nTiles);
}